// FrozenBNBStableEmbedding_76295799046276
// MI455X (gfx1250) — compile-verified
//
#include <hip/hip_runtime.h>
#include <hip/hip_bf16.h>

#ifndef __has_builtin
#define __has_builtin(x) 0
#endif

#if __has_builtin(__builtin_amdgcn_global_load_async_to_lds_b128)
#define HAVE_ASYNC_LDS 1
#else
#define HAVE_ASYNC_LDS 0
#endif

#define EPS    1e-5f
#define DIM    1024
#define TPB    8      // tokens per block: one token per wave32

typedef int v4i __attribute__((ext_vector_type(4)));
typedef __attribute__((address_space(1))) v4i* gptr_v4i;
typedef __attribute__((address_space(3))) v4i* lptr_v4i;

// One wave32 per token. Lane L owns elements d = c*128 + L*4 .. +3 for c=0..7
// -> every global b128 access is a contiguous 512B span across the wave.
__global__ __launch_bounds__(256) void embln_kernel(
    const int*   __restrict__ x,       // [ntok] token ids
    const int*   __restrict__ w8,      // [V, DIM] int32 codes 0..255
    const float* __restrict__ absmax,  // [V*DIM/4096]
    const float* __restrict__ code,    // [256]
    const float* __restrict__ lnw,     // [DIM]
    const float* __restrict__ lnb,     // [DIM]
    float*       __restrict__ out,     // [ntok, DIM]
    int ntok)
{
    __shared__ __align__(16) float s_code[256];
    __shared__ __align__(16) float s_w[DIM];
    __shared__ __align__(16) float s_b[DIM];
    __shared__ __align__(16) float s_row[TPB][DIM];  // raw codes, then dequant floats

    const int tid  = threadIdx.x;
    const int lane = tid & 31;
    const int wv   = tid >> 5;

    // Cooperative table load (256 threads): code LUT + LN affine params.
    s_code[tid] = code[tid];
#pragma unroll
    for (int i = 0; i < 4; ++i) {
        s_w[tid + 256 * i] = lnw[tid + 256 * i];
        s_b[tid + 256 * i] = lnb[tid + 256 * i];
    }

    const int  t      = blockIdx.x * TPB + wv;
    const bool active = (t < ntok);

    int   row   = 0;
    float scale = 0.f;
    if (active) {
        row = x[t];
        // BLOCK = 4096 = 4 rows of DIM=1024; a row never crosses a block.
        scale = absmax[row >> 2];
    }

#if HAVE_ASYNC_LDS
    // Async DMA the 4KB row into this wave's LDS slot (8 x b128 per lane,
    // ASYNCcnt-tracked) while the rest of the block finishes table setup.
    if (active) {
        int* gsrc = (int*)(w8 + (size_t)row * DIM + lane * 4); // drop const for builtin
        int* ldst = (int*)&s_row[wv][0] + lane * 4;
#define ASYNC_CP(OFF) \
        __builtin_amdgcn_global_load_async_to_lds_b128( \
            (gptr_v4i)gsrc, (lptr_v4i)ldst, (OFF), /*cpol=*/0)
        ASYNC_CP(0);
        ASYNC_CP(512);
        ASYNC_CP(1024);
        ASYNC_CP(1536);
        ASYNC_CP(2048);
        ASYNC_CP(2560);
        ASYNC_CP(3072);
        ASYNC_CP(3584);
#undef ASYNC_CP
    }
#endif

    __syncthreads();  // tables visible to all waves

    if (active) {
#if HAVE_ASYNC_LDS
#if __has_builtin(__builtin_amdgcn_s_wait_asynccnt)
        __builtin_amdgcn_s_wait_asynccnt(0);
#else
        asm volatile("s_wait_asynccnt 0" ::: "memory");
#endif
#endif
        // Pass 1: dequantize via LDS LUT, accumulate sum / sumsq, stash floats.
        float sum = 0.f, ssq = 0.f;
#pragma unroll
        for (int c = 0; c < 8; ++c) {
            const int d = c * 128 + lane * 4;
#if HAVE_ASYNC_LDS
            int4 q = *(const int4*)((const int*)&s_row[wv][0] + d);
#else
            int4 q = *(const int4*)(w8 + (size_t)row * DIM + d);
#endif
            float4 v;
            v.x = s_code[q.x & 255] * scale;
            v.y = s_code[q.y & 255] * scale;
            v.z = s_code[q.z & 255] * scale;
            v.w = s_code[q.w & 255] * scale;
            sum += (v.x + v.y) + (v.z + v.w);
            ssq += (v.x * v.x + v.y * v.y) + (v.z * v.z + v.w * v.w);
            *(float4*)&s_row[wv][d] = v;   // overwrite codes with dequant values
        }

        // wave32 butterfly reduction
#pragma unroll
        for (int off = 16; off >= 1; off >>= 1) {
            sum += __shfl_xor(sum, off, 32);
            ssq += __shfl_xor(ssq, off, 32);
        }
        const float mean = sum * (1.f / DIM);
        const float var  = ssq * (1.f / DIM) - mean * mean;
        const float rstd = rsqrtf(var + EPS);

        // Pass 2: normalize + affine, coalesced float4 stores.
        float* po = out + (size_t)t * DIM;
#pragma unroll
        for (int c = 0; c < 8; ++c) {
            const int d = c * 128 + lane * 4;
            float4 v = *(const float4*)&s_row[wv][d];
            float4 w = *(const float4*)&s_w[d];
            float4 b = *(const float4*)&s_b[d];
            float4 o;
            o.x = (v.x - mean) * rstd * w.x + b.x;
            o.y = (v.y - mean) * rstd * w.y + b.y;
            o.z = (v.z - mean) * rstd * w.z + b.z;
            o.w = (v.w - mean) * rstd * w.w + b.w;
            *(float4*)(po + d) = o;
        }
    }
}

extern "C" void kernel_launch(void* const* d_in, const int* in_sizes, int n_in,
                              void* d_out, int out_size, void* d_ws, size_t ws_size,
                              hipStream_t stream) {
    const int*   x      = (const int*)  d_in[0];  // token ids
    const int*   w8     = (const int*)  d_in[1];  // [V, D] codes
    const float* absmax = (const float*)d_in[2];
    const float* code   = (const float*)d_in[3];
    const float* lnw    = (const float*)d_in[4];
    const float* lnb    = (const float*)d_in[5];
    float*       out    = (float*)d_out;

    const int ntok    = in_sizes[0];            // 8*2048 = 16384
    const int nblocks = (ntok + TPB - 1) / TPB; // 2048

    embln_kernel<<<nblocks, 256, 0, stream>>>(x, w8, absmax, code, lnw, lnb, out, ntok);
}